// GlobalAttentionNet_38371237822820
// MI455X (gfx1250) — compile-verified
//
#include <hip/hip_runtime.h>
#include <hip/hip_bf16.h>
#include <math.h>

// ---------------------------------------------------------------------------
// GlobalAttentionNet for MI455X (gfx1250, wave32).
// GEMMs use V_WMMA_F32_16X16X4_F32 (full fp32 fidelity); graph aggregation is
// L2-resident float atomics (memory-bound part of the roofline).
// Degree normalization is hoisted out of the WMMA K-loop (applied per output
// row in the epilogue), and the Wl/Wr GEMMs run as two interleaved
// accumulator chains so WMMA RAW hazards are hidden by alternate issue.
// ---------------------------------------------------------------------------

typedef float v2f __attribute__((ext_vector_type(2)));
typedef float v8f __attribute__((ext_vector_type(8)));

#define HDIM 128

// ---------------------------------------------------------------------------
// Edge scatter: agg[dst] += x[src], deg[dst] += 1.  32 threads per edge,
// each thread handles 4 contiguous floats (float4 gather, 4 f32 atomics).
// ---------------------------------------------------------------------------
__global__ __launch_bounds__(256) void scatter_kernel(
    const float* __restrict__ x,
    const long long* __restrict__ src,
    const long long* __restrict__ dst,
    float* __restrict__ agg,
    float* __restrict__ deg,
    long long nedges)
{
    long long gid  = (long long)blockIdx.x * blockDim.x + threadIdx.x;
    long long e    = gid >> 5;
    int       part = (int)(gid & 31);
    if (e >= nedges) return;
    long long s = src[e];
    long long d = dst[e];
    const float4 v = *(const float4*)(x + s * HDIM + part * 4);
    float* a = agg + d * HDIM + part * 4;
    atomicAdd(a + 0, v.x);
    atomicAdd(a + 1, v.y);
    atomicAdd(a + 2, v.z);
    atomicAdd(a + 3, v.w);
    if (part == 0) atomicAdd(deg + d, 1.0f);
}

// ---------------------------------------------------------------------------
// Fused SAGE layer: out = relu( (agg @ Wl)/max(deg,1) + b + x @ Wr )
// Block = 256 threads = 8 waves; block computes rows [row0,row0+16) x all 128
// cols; wave w computes the 16x16 tile at cols [16w,16w+16).
// K-loop in steps of 4 via V_WMMA_F32_16X16X4_F32, two accumulator chains.
//   A (16x4 f32): lane(0-15)=M, vgpr0/1 = K {0,1}; lanes 16-31 = K {2,3}.
//   B (4x16 f32): lane%16 = N, vgpr0/1 = K {0,1} (lanes 16-31 K {2,3}).
//   C/D (16x16):  vgpr j -> row j (+8 for lanes 16-31), lane%16 = N.
// Grid must be exact (N % 16 == 0) so EXEC is all ones for WMMA.
// ---------------------------------------------------------------------------
__global__ __launch_bounds__(256) void sage_gemm_kernel(
    const float* __restrict__ agg,
    const float* __restrict__ deg,
    const float* __restrict__ xin,
    const float* __restrict__ Wl,
    const float* __restrict__ bias,
    const float* __restrict__ Wr,
    float* __restrict__ out)
{
    const int row0 = blockIdx.x * 16;
    const int wave = threadIdx.x >> 5;   // 0..7
    const int lane = threadIdx.x & 31;
    const int col0 = wave * 16;
    const int m    = lane & 15;          // M for A, N for B/D
    const int kb   = (lane >> 4) << 1;   // 0 or 2

    const float* aRow = agg + (size_t)(row0 + m) * HDIM;
    const float* xRow = xin + (size_t)(row0 + m) * HDIM;

    v8f accL = {};   // agg @ Wl
    v8f accR = {};   // x   @ Wr

    #pragma unroll 4
    for (int kk = 0; kk < HDIM; kk += 4) {
        v2f a  = *(const v2f*)(aRow + kk + kb);
        v2f xa = *(const v2f*)(xRow + kk + kb);
        v2f bl, br;
        bl.x = Wl[(size_t)(kk + kb) * HDIM + col0 + m];
        bl.y = Wl[(size_t)(kk + kb + 1) * HDIM + col0 + m];
        br.x = Wr[(size_t)(kk + kb) * HDIM + col0 + m];
        br.y = Wr[(size_t)(kk + kb + 1) * HDIM + col0 + m];
        accL = __builtin_amdgcn_wmma_f32_16x16x4_f32(
            false, a,  false, bl, (short)0, accL, false, false);
        accR = __builtin_amdgcn_wmma_f32_16x16x4_f32(
            false, xa, false, br, (short)0, accR, false, false);
    }

    // Epilogue: acc[j] belongs to output row rbase+j -> normalize per that row.
    const float bv   = bias[col0 + m];
    const int  rbase = row0 + ((lane >> 4) << 3);   // +8 row offset for hi lanes
    #pragma unroll
    for (int j = 0; j < 8; ++j) {
        float invd = 1.0f / fmaxf(deg[rbase + j], 1.0f);
        float v = accL[j] * invd + accR[j] + bv;
        out[(size_t)(rbase + j) * HDIM + col0 + m] = fmaxf(v, 0.0f);
    }
}

// ---------------------------------------------------------------------------
// Attention gate: gate[i] = h[i,:] . gw + gb ; segment max via float atomics.
// One wave per node.
// ---------------------------------------------------------------------------
__device__ __forceinline__ void atomicMaxFloat(float* addr, float value)
{
    if (value >= 0.0f)
        atomicMax((int*)addr, __float_as_int(value));
    else
        atomicMin((unsigned int*)addr, __float_as_uint(value));
}

__global__ __launch_bounds__(256) void gate_kernel(
    const float* __restrict__ h,
    const float* __restrict__ gw,
    const float* __restrict__ gb,
    const long long* __restrict__ batch,
    float* __restrict__ gate,
    float* __restrict__ gmax,
    int n)
{
    int wid  = (int)(((long long)blockIdx.x * blockDim.x + threadIdx.x) >> 5);
    int lane = threadIdx.x & 31;
    if (wid >= n) return;
    const float* row = h + (size_t)wid * HDIM;
    float s = 0.0f;
    #pragma unroll
    for (int k = lane; k < HDIM; k += 32) s += row[k] * gw[k];
    #pragma unroll
    for (int off = 16; off > 0; off >>= 1) s += __shfl_xor(s, off, 32);
    if (lane == 0) {
        s += gb[0];
        gate[wid] = s;
        atomicMaxFloat(&gmax[batch[wid]], s);
    }
}

__global__ __launch_bounds__(256) void init_gmax_kernel(float* gmax, int g)
{
    int i = blockIdx.x * blockDim.x + threadIdx.x;
    if (i < g) gmax[i] = -INFINITY;
}

// e[i] = exp(gate[i] - gmax[batch[i]]); denom[batch[i]] += e[i]
__global__ __launch_bounds__(256) void softmax_num_kernel(
    float* __restrict__ gate,
    const float* __restrict__ gmax,
    const long long* __restrict__ batch,
    float* __restrict__ denom,
    int n)
{
    int i = (int)((long long)blockIdx.x * blockDim.x + threadIdx.x);
    if (i >= n) return;
    long long g = batch[i];
    float e = __expf(gate[i] - gmax[g]);
    gate[i] = e;
    atomicAdd(&denom[g], e);
}

// pooled[batch[i], f] += h[i,f] * e[i]/denom[batch[i]]; thread per (node,f)
__global__ __launch_bounds__(256) void pool_kernel(
    const float* __restrict__ h,
    const float* __restrict__ gate,
    const float* __restrict__ denom,
    const long long* __restrict__ batch,
    float* __restrict__ pooled,
    long long total)
{
    long long gid = (long long)blockIdx.x * blockDim.x + threadIdx.x;
    if (gid >= total) return;
    long long i = gid >> 7;          // node
    int f = (int)(gid & (HDIM - 1)); // feature
    long long g = batch[i];
    float alpha = gate[i] / denom[g];
    atomicAdd(&pooled[g * HDIM + f], h[i * HDIM + f] * alpha);
}

// ---------------------------------------------------------------------------
// Head: out = sigmoid(relu(pooled @ W1 + b1) @ W2 + b2). Block per graph.
// ---------------------------------------------------------------------------
__global__ __launch_bounds__(128) void head_kernel(
    const float* __restrict__ pooled,
    const float* __restrict__ w1, const float* __restrict__ b1,
    const float* __restrict__ w2, const float* __restrict__ b2,
    float* __restrict__ out)
{
    __shared__ float row[HDIM];
    __shared__ float hid[HDIM];
    int g = blockIdx.x, t = threadIdx.x;
    row[t] = pooled[(size_t)g * HDIM + t];
    __syncthreads();
    float acc = b1[t];
    #pragma unroll 8
    for (int k = 0; k < HDIM; ++k) acc += row[k] * w1[(size_t)k * HDIM + t];
    hid[t] = fmaxf(acc, 0.0f);
    __syncthreads();
    if (t < 2) {
        float a = b2[t];
        #pragma unroll 8
        for (int k = 0; k < HDIM; ++k) a += hid[k] * w2[k * 2 + t];
        out[g * 2 + t] = 1.0f / (1.0f + __expf(-a));
    }
}

// ---------------------------------------------------------------------------
extern "C" void kernel_launch(void* const* d_in, const int* in_sizes, int n_in,
                              void* d_out, int out_size, void* d_ws, size_t ws_size,
                              hipStream_t stream)
{
    const float*     x      = (const float*)d_in[0];
    const long long* eidx   = (const long long*)d_in[1];
    const long long* batch  = (const long long*)d_in[2];
    const float*     w1_l   = (const float*)d_in[3];
    const float*     b1     = (const float*)d_in[4];
    const float*     w1_r   = (const float*)d_in[5];
    const float*     ws_l   = (const float*)d_in[6];   // [2,128,128]
    const float*     bs     = (const float*)d_in[7];   // [2,128]
    const float*     ws_r   = (const float*)d_in[8];
    const float*     gate_w = (const float*)d_in[9];
    const float*     gate_b = (const float*)d_in[10];
    const float*     lin1_w = (const float*)d_in[11];
    const float*     lin1_b = (const float*)d_in[12];
    const float*     lin2_w = (const float*)d_in[13];
    const float*     lin2_b = (const float*)d_in[14];
    float*           outp   = (float*)d_out;

    const int       N = in_sizes[0] / HDIM;           // 100000 (divisible by 16)
    const long long E = (long long)in_sizes[1] / 2;   // 1.6M
    const int       G = out_size / 2;                 // 512

    const long long* src = eidx;
    const long long* dst = eidx + E;

    // Workspace layout (floats)
    float* ws = (float*)d_ws;
    size_t p = 0;
    float* agg    = ws + p; p += (size_t)N * HDIM;
    float* hA     = ws + p; p += (size_t)N * HDIM;
    float* hB     = ws + p; p += (size_t)N * HDIM;
    float* deg    = ws + p; p += (size_t)N;
    float* gate   = ws + p; p += (size_t)N;
    float* gmax   = ws + p; p += (size_t)G;
    float* denom  = ws + p; p += (size_t)G;
    float* pooled = ws + p; p += (size_t)G * HDIM;

    const int scatterBlocks = (int)((E * 32 + 255) / 256);
    const int gemmBlocks    = N / 16;        // N % 16 == 0 -> full EXEC for WMMA

    struct Layer { const float *xin, *wl, *b, *wr; float* hout; };
    Layer layers[3] = {
        { x,  w1_l,              b1,        w1_r,              hA },
        { hA, ws_l,              bs,        ws_r,              hB },
        { hB, ws_l + HDIM*HDIM,  bs + HDIM, ws_r + HDIM*HDIM,  hA },
    };

    for (int l = 0; l < 3; ++l) {
        hipMemsetAsync(agg, 0, (size_t)N * HDIM * sizeof(float), stream);
        hipMemsetAsync(deg, 0, (size_t)N * sizeof(float), stream);
        scatter_kernel<<<scatterBlocks, 256, 0, stream>>>(
            layers[l].xin, src, dst, agg, deg, E);
        sage_gemm_kernel<<<gemmBlocks, 256, 0, stream>>>(
            agg, deg, layers[l].xin, layers[l].wl, layers[l].b, layers[l].wr,
            layers[l].hout);
    }
    float* h = hA;   // final node embeddings

    // Attentional aggregation
    hipMemsetAsync(denom,  0, (size_t)G * sizeof(float), stream);
    hipMemsetAsync(pooled, 0, (size_t)G * HDIM * sizeof(float), stream);
    init_gmax_kernel<<<(G + 255) / 256, 256, 0, stream>>>(gmax, G);

    gate_kernel<<<((long long)N * 32 + 255) / 256, 256, 0, stream>>>(
        h, gate_w, gate_b, batch, gate, gmax, N);
    softmax_num_kernel<<<(N + 255) / 256, 256, 0, stream>>>(
        gate, gmax, batch, denom, N);
    pool_kernel<<<(int)(((long long)N * HDIM + 255) / 256), 256, 0, stream>>>(
        h, gate, denom, batch, pooled, (long long)N * HDIM);

    // MLP head + sigmoid
    head_kernel<<<G, 128, 0, stream>>>(
        pooled, lin1_w, lin1_b, lin2_w, lin2_b, outp);
}